// BayesianLayer_78606491451970
// MI455X (gfx1250) — compile-verified
//
#include <hip/hip_runtime.h>

typedef float v2f __attribute__((ext_vector_type(2)));
typedef float v8f __attribute__((ext_vector_type(8)));
typedef unsigned int u32x4 __attribute__((ext_vector_type(4)));
typedef int i32x8 __attribute__((ext_vector_type(8)));
typedef int i32x4 __attribute__((ext_vector_type(4)));

#define BATCH 128
#define IN    1024
#define OUT   1024

// fast softplus using hw transcendentals: log(1+exp(-|v|)) + max(v,0)
// (v_exp_f32 + v_add + v_log_f32 + v_max, ~5 VALU ops vs ~30 for log1pf)
__device__ __forceinline__ float softplus_f(float v) {
    return __logf(1.0f + __expf(-fabsf(v))) + fmaxf(v, 0.0f);
}

// Block: 128 threads = 4 waves (wave32).
// Each block: 16 batch rows x 64 out cols. Each wave: one 16x16 WMMA tile.
// out[b,o] = sum_i x[b,i]*mu[i,o]                (WMMA f32 16x16x4 over K=IN)
//          + sum_i x[b,i]*eps[b,i,o]*sigma[i,o]  (VALU FMAs, eps streamed once)
//          + eps_bias[b,o]*softplus(ro_bias[o]) + mu_bias[o]
__global__ __launch_bounds__(128)
void bayes_fused_kernel(const float* __restrict__ x,
                        const float* __restrict__ eps,
                        const float* __restrict__ eps_bias,
                        const float* __restrict__ mu,
                        const float* __restrict__ ro,
                        const float* __restrict__ mu_bias,
                        const float* __restrict__ ro_bias,
                        float* __restrict__ out) {
    __shared__ float xs[16 * IN];  // 64 KB: x tile [16 batch][IN], row-major

    const int tid  = threadIdx.x;      // 0..127
    const int wave = tid >> 5;         // 0..3
    const int lane = tid & 31;         // 0..31
    const int b0   = blockIdx.y * 16;              // batch tile origin
    const int o0   = blockIdx.x * 64 + wave * 16;  // out tile origin (per wave)

    // ---- stage x[b0:b0+16, :] into LDS via Tensor Data Mover (one TDM op) ----
    if (wave == 0) {
        const unsigned long long ga =
            (unsigned long long)(uintptr_t)(x + (size_t)b0 * IN);
        const unsigned int lds = (unsigned int)(uintptr_t)&xs[0];  // LDS byte offset

        // D# group 0 (128b): count=1, is_store=0, lds_addr, global_addr, type=2
        u32x4 g0;
        g0.x = 1u;                                   // count=1, user mode
        g0.y = lds;                                  // bits[63:32] lds_addr
        g0.z = (unsigned int)ga;                     // global_addr[31:0]
        g0.w = (unsigned int)(ga >> 32) | 0x80000000u;  // ga[56:32] | type=2<<30

        // D# group 1 (256b): data_size=4B, 2-D tensor 1024x128, tile 1024x16,
        // dim0 stride = 1024 elements.
        i32x8 g1;
        g1[0] = 0x00020000;          // workgroup_mask=0 | data_size(2=4B)<<16
        g1[1] = (int)(1024u << 16);  // tensor_dim0[15:0]=1024 in bits[63:48]
        g1[2] = (int)(128u << 16);   // dim0[31:16]=0 | tensor_dim1[15:0]=128
        g1[3] = (int)(1024u << 16);  // dim1[31:16]=0 | tile_dim0=1024
        g1[4] = 16;                  // tile_dim1=16 | tile_dim2=0
        g1[5] = 1024;                // tensor_dim0_stride[31:0]=1024
        g1[6] = 0;                   // stride0[47:32]=0 | stride1[15:0]=0
        g1[7] = 0;                   // stride1[47:16]=0

        const i32x4 gz4 = {0, 0, 0, 0};              // groups 2/3 unused (2-D)
        const i32x8 gz8 = {0, 0, 0, 0, 0, 0, 0, 0};  // extra group (clang-23 form)
        __builtin_amdgcn_tensor_load_to_lds(g0, g1, gz4, gz4, gz8, /*cpol=*/0);
        __builtin_amdgcn_s_wait_tensorcnt(0);
    }
    __syncthreads();

    // WMMA lane mapping (wave32):
    //  C/D: VGPR r -> row m = r + 8*(lane>=16), col n = lane&15
    //  A 16x4: lanes 0-15 hold K=0,1 ; lanes 16-31 hold K=2,3 ; row m = lane&15
    //  B 4x16: lanes 0-15 hold K=0,1 ; lanes 16-31 hold K=2,3 ; col n = lane&15
    const int n   = lane & 15;
    const int hi  = lane >> 4;     // 0 or 1
    const int k0  = hi * 2;        // K offset carried by upper half-wave
    const int o   = o0 + n;

    v8f   c = {0.f, 0.f, 0.f, 0.f, 0.f, 0.f, 0.f, 0.f};
    float acc[8] = {0.f, 0.f, 0.f, 0.f, 0.f, 0.f, 0.f, 0.f};

    const float* mu_col = mu + o;
    const float* ro_col = ro + o;
    // per-lane eps column base for this block's batch tile
    const float* eps_col = eps + ((size_t)b0 * IN) * OUT + o;

    for (int i0 = 0; i0 < IN; i0 += 4) {
        // ---- WMMA part: c += x_tile(16x4) * mu_tile(4x16) ----
        v2f a, bfrag;
        {
            // A fragment from LDS: row = lane&15, K = i0+k0, i0+k0+1
            const float* xr = &xs[(lane & 15) * IN + i0 + k0];
            a.x = xr[0];
            a.y = xr[1];
            // B fragment from mu (L2-resident): rows i0+k0, i0+k0+1, col o
            bfrag.x = mu_col[(size_t)(i0 + k0) * OUT];
            bfrag.y = mu_col[(size_t)(i0 + k0 + 1) * OUT];
        }
        c = __builtin_amdgcn_wmma_f32_16x16x4_f32(
                /*neg_a=*/false, a, /*neg_b=*/false, bfrag,
                /*c_mod=*/(short)0, c, /*reuse_a=*/false, /*reuse_b=*/false);

        // ---- eps part: sigma for the 4 K-slices, then 16x16 FMAs ----
        const float s0 = softplus_f(ro_col[(size_t)(i0 + 0) * OUT]);
        const float s1 = softplus_f(ro_col[(size_t)(i0 + 1) * OUT]);
        const float s2 = softplus_f(ro_col[(size_t)(i0 + 2) * OUT]);
        const float s3 = softplus_f(ro_col[(size_t)(i0 + 3) * OUT]);

        #pragma unroll
        for (int r = 0; r < 8; ++r) {
            const int m = r + hi * 8;                       // batch row in tile
            const float4 xv = *(const float4*)&xs[m * IN + i0];  // ds_load_b128
            const float* ep = eps_col + ((size_t)m * IN + i0) * OUT;
            if (i0 + 16 < IN) __builtin_prefetch(ep + 16 * OUT, 0, 3);  // prime HBM stream
            acc[r] += xv.x * ep[0 * OUT] * s0;
            acc[r] += xv.y * ep[1 * OUT] * s1;
            acc[r] += xv.z * ep[2 * OUT] * s2;
            acc[r] += xv.w * ep[3 * OUT] * s3;
        }
    }

    // ---- epilogue: bias + writeout (matches C/D VGPR layout) ----
    const float sb = softplus_f(ro_bias[o]);
    const float mb = mu_bias[o];
    #pragma unroll
    for (int r = 0; r < 8; ++r) {
        const int b = b0 + r + hi * 8;
        const float bias = eps_bias[(size_t)b * OUT + o] * sb + mb;
        out[(size_t)b * OUT + o] = c[r] + acc[r] + bias;
    }
}

extern "C" void kernel_launch(void* const* d_in, const int* in_sizes, int n_in,
                              void* d_out, int out_size, void* d_ws, size_t ws_size,
                              hipStream_t stream) {
    (void)in_sizes; (void)n_in; (void)d_ws; (void)ws_size; (void)out_size;
    const float* x        = (const float*)d_in[0];
    const float* eps      = (const float*)d_in[1];
    const float* eps_bias = (const float*)d_in[2];
    const float* mu       = (const float*)d_in[3];
    const float* ro       = (const float*)d_in[4];
    const float* mu_bias  = (const float*)d_in[5];
    const float* ro_bias  = (const float*)d_in[6];
    float* out = (float*)d_out;

    dim3 grid(OUT / 64, BATCH / 16);  // 16 x 8 = 128 workgroups
    dim3 block(128);                  // 4 waves of 32
    bayes_fused_kernel<<<grid, block, 0, stream>>>(x, eps, eps_bias, mu, ro,
                                                   mu_bias, ro_bias, out);
}